// GNLoss_23098334118301
// MI455X (gfx1250) — compile-verified
//
#include <hip/hip_runtime.h>

// GN loss for B=8, C=128, H=128, W=256, N=1024 (hardcoded from reference).
// Strategy: one wave = 16 points; 5 chained V_WMMA_F32_16X16X4_F32 Gram
// products accumulate the per-point channel reductions on the D diagonal.
// Gradients computed on the fly from a 12-point fb stencil (no gx/gy arrays).

typedef __attribute__((ext_vector_type(2))) float v2f;
typedef __attribute__((ext_vector_type(8))) float v8f;

#define GN_B 8
#define GN_C 128
#define GN_H 128
#define GN_W 256
#define GN_N 1024
#define GN_HW (GN_H * GN_W)
#define GN_PTS (GN_B * GN_N)   // 8192
#define GN_WAVES (GN_PTS / 16) // 512 waves, 16 points each

__device__ __forceinline__ float sel8(v8f v, int i) {
  float r = v[0];
  r = (i == 1) ? v[1] : r;
  r = (i == 2) ? v[2] : r;
  r = (i == 3) ? v[3] : r;
  r = (i == 4) ? v[4] : r;
  r = (i == 5) ? v[5] : r;
  r = (i == 6) ? v[6] : r;
  r = (i == 7) ? v[7] : r;
  return r;
}

struct Geom {
  int o_ym_x0, o_ym_x1;
  int o_y0_xm, o_y0_x0, o_y0_x1, o_y0_xp;
  int o_y1_xm, o_y1_x0, o_y1_x1, o_y1_xp;
  int o_yp_x0, o_yp_x1;
  float w00, w01, w10, w11;
  float sx0, sx1, sy0, sy1; // gradient scales (0.5 interior, 1.0 at edges)
};

// Bilinear sample of f, gx, gy at one channel; jnp.gradient edge handling baked in.
__device__ __forceinline__ void sample3(const float* __restrict__ bp, const Geom& g,
                                        float& fs, float& jx, float& jy) {
  float fym0 = bp[g.o_ym_x0];
  float fym1 = bp[g.o_ym_x1];
  float f0m  = bp[g.o_y0_xm];
  float f00  = bp[g.o_y0_x0];
  float f01  = bp[g.o_y0_x1];
  float f0p  = bp[g.o_y0_xp];
  float f1m  = bp[g.o_y1_xm];
  float f10  = bp[g.o_y1_x0];
  float f11  = bp[g.o_y1_x1];
  float f1p  = bp[g.o_y1_xp];
  float fyp0 = bp[g.o_yp_x0];
  float fyp1 = bp[g.o_yp_x1];

  float gx00 = (f01 - f0m) * g.sx0;
  float gx01 = (f0p - f00) * g.sx1;
  float gx10 = (f11 - f1m) * g.sx0;
  float gx11 = (f1p - f10) * g.sx1;
  float gy00 = (f10 - fym0) * g.sy0;
  float gy01 = (f11 - fym1) * g.sy0;
  float gy10 = (fyp0 - f00) * g.sy1;
  float gy11 = (fyp1 - f10) * g.sy1;

  fs = g.w00 * f00 + g.w01 * f01 + g.w10 * f10 + g.w11 * f11;
  jx = g.w00 * gx00 + g.w01 * gx01 + g.w10 * gx10 + g.w11 * gx11;
  jy = g.w00 * gy00 + g.w01 * gy01 + g.w10 * gy10 + g.w11 * gy11;
}

__global__ __launch_bounds__(256) void gn_main(const float* __restrict__ fb,
                                               const float* __restrict__ ft,
                                               const float* __restrict__ ub,
                                               const float* __restrict__ noise,
                                               float* __restrict__ ws) {
  const int lane = threadIdx.x & 31;
  const int wave = blockIdx.x * (blockDim.x >> 5) + (threadIdx.x >> 5);
  const int p    = lane & 15;          // point within the wave's 16-point group
  const int n    = wave * 16 + p;      // global point id in [0, 8192)
  const int b    = n >> 10;            // n / N

  // ---- per-point geometry (computed redundantly in both lane halves) ----
  float ubx = ub[2 * n + 0], uby = ub[2 * n + 1];
  float xsx = ubx + noise[2 * n + 0];
  float xsy = uby + noise[2 * n + 1];
  float xc = fminf(fmaxf(xsx, 0.0f), (float)(GN_W - 1));
  float yc = fminf(fmaxf(xsy, 0.0f), (float)(GN_H - 1));
  int x0 = (int)floorf(xc); x0 = x0 < 0 ? 0 : (x0 > GN_W - 2 ? GN_W - 2 : x0);
  int y0 = (int)floorf(yc); y0 = y0 < 0 ? 0 : (y0 > GN_H - 2 ? GN_H - 2 : y0);
  const int x1 = x0 + 1, y1 = y0 + 1;
  const float wx = xc - (float)x0, wy = yc - (float)y0;

  Geom g;
  g.w00 = (1.0f - wx) * (1.0f - wy);
  g.w01 = wx * (1.0f - wy);
  g.w10 = (1.0f - wx) * wy;
  g.w11 = wx * wy;
  const int xm = x0 > 0 ? x0 - 1 : 0;
  const int xp = x1 < GN_W - 1 ? x1 + 1 : GN_W - 1;
  const int ym = y0 > 0 ? y0 - 1 : 0;
  const int yp = y1 < GN_H - 1 ? y1 + 1 : GN_H - 1;
  g.sx0 = (x0 == 0)        ? 1.0f : 0.5f;
  g.sx1 = (x1 == GN_W - 1) ? 1.0f : 0.5f;
  g.sy0 = (y0 == 0)        ? 1.0f : 0.5f;
  g.sy1 = (y1 == GN_H - 1) ? 1.0f : 0.5f;
  g.o_ym_x0 = ym * GN_W + x0;  g.o_ym_x1 = ym * GN_W + x1;
  g.o_y0_xm = y0 * GN_W + xm;  g.o_y0_x0 = y0 * GN_W + x0;
  g.o_y0_x1 = y0 * GN_W + x1;  g.o_y0_xp = y0 * GN_W + xp;
  g.o_y1_xm = y1 * GN_W + xm;  g.o_y1_x0 = y1 * GN_W + x0;
  g.o_y1_x1 = y1 * GN_W + x1;  g.o_y1_xp = y1 * GN_W + xp;
  g.o_yp_x0 = yp * GN_W + x0;  g.o_yp_x1 = yp * GN_W + x1;

  const float* fbBase = fb + (size_t)b * GN_C * GN_HW;
  const float* ftRow  = ft + (size_t)n * GN_C;

  // ---- Gram accumulation via f32 WMMA 16x16x4 ----
  // A-layout (32-bit 16x4): lane L -> row M=L%16; K slots {0,1} (L<16) / {2,3} (L>=16),
  // VGPR0 = first slot, VGPR1 = second slot. B mirrors this; diagonals of D
  // give the per-point channel sums.
  const int slot = (lane >= 16) ? 2 : 0;
  v8f accXX = {}, accXY = {}, accYY = {}, accXR = {}, accYR = {};

  for (int kb = 0; kb < GN_C / 4; ++kb) {
    const int ca = kb * 4 + slot;
    float fsa, jxa, jya, fsb, jxb, jyb;
    sample3(fbBase + (size_t)ca * GN_HW, g, fsa, jxa, jya);
    sample3(fbBase + (size_t)(ca + 1) * GN_HW, g, fsb, jxb, jyb);
    float ra = fsa - ftRow[ca];
    float rb = fsb - ftRow[ca + 1];
    v2f aJx = {jxa, jxb};
    v2f aJy = {jya, jyb};
    v2f aR  = {ra,  rb};
    accXX = __builtin_amdgcn_wmma_f32_16x16x4_f32(false, aJx, false, aJx, (short)0, accXX, false, false);
    accXY = __builtin_amdgcn_wmma_f32_16x16x4_f32(false, aJx, false, aJy, (short)0, accXY, false, false);
    accYY = __builtin_amdgcn_wmma_f32_16x16x4_f32(false, aJy, false, aJy, (short)0, accYY, false, false);
    accXR = __builtin_amdgcn_wmma_f32_16x16x4_f32(false, aJx, false, aR,  (short)0, accXR, false, false);
    accYR = __builtin_amdgcn_wmma_f32_16x16x4_f32(false, aJy, false, aR,  (short)0, accYR, false, false);
  }

  // ---- extract diagonal: element (n,n) -> VGPR n%8, lane n (n<8) / n+16 (n>=8) ----
  const bool owner = (lane < 8) || (lane >= 24);
  const int vi = lane & 7;
  float sxx = sel8(accXX, vi);
  float sxy = sel8(accXY, vi);
  float syy = sel8(accYY, vi);
  float bx  = sel8(accXR, vi);
  float by  = sel8(accYR, vi);

  // 2x2 GN solve (computed in all lanes, masked by select — no divergence)
  float a_ = sxx + 1e-9f;
  float d_ = syy + 1e-9f;
  float bb = sxy;
  float det = a_ * d_ - bb * bb;
  float inv = 1.0f / det;
  float hbx = ( d_ * bx - bb * by) * inv;   // (Hinv @ b).x
  float hby = (-bb * bx + a_ * by) * inv;   // (Hinv @ b).y
  float dx = ubx - xsx + hbx;               // diff = ub - (xs - Hinv b)
  float dy = uby - xsy + hby;
  float quad = a_ * dx * dx + 2.0f * bb * dx * dy + d_ * dy * dy;
  float ldet = logf(fmaxf(det, 1e-16f));
  quad = owner ? quad : 0.0f;
  ldet = owner ? ldet : 0.0f;

  // wave32 reduction of the two scalars
  for (int off = 16; off > 0; off >>= 1) {
    quad += __shfl_xor(quad, off, 32);
    ldet += __shfl_xor(ldet, off, 32);
  }
  if (lane == 0) {
    ws[2 * wave + 0] = quad;
    ws[2 * wave + 1] = ldet;
  }
}

__global__ void gn_finalize(const float* __restrict__ ws, float* __restrict__ out) {
  if (threadIdx.x == 0) {
    float sq = 0.0f, sl = 0.0f;
    for (int i = 0; i < GN_WAVES; ++i) {
      sq += ws[2 * i + 0];
      sl += ws[2 * i + 1];
    }
    float e1 = 0.5f * sq;
    float e2 = (float)GN_PTS * 1.8378770664093453f - 0.5f * sl; // BN*log(2*pi) - 0.5*sum(log det)
    float e  = 1.0f * e1 + (2.0f / 7.0f) * e2;
    out[0] = e;
    out[1] = e1;
    out[2] = e2;
  }
}

extern "C" void kernel_launch(void* const* d_in, const int* in_sizes, int n_in,
                              void* d_out, int out_size, void* d_ws, size_t ws_size,
                              hipStream_t stream) {
  (void)in_sizes; (void)n_in; (void)out_size; (void)ws_size;
  const float* fb    = (const float*)d_in[0];
  const float* ft    = (const float*)d_in[1];
  const float* ub    = (const float*)d_in[2];
  const float* noise = (const float*)d_in[3];
  float* ws  = (float*)d_ws;
  float* out = (float*)d_out;

  // 8192 points / 16 per wave = 512 waves = 64 blocks of 8 waves (256 thr).
  gn_main<<<64, 256, 0, stream>>>(fb, ft, ub, noise, ws);
  gn_finalize<<<1, 32, 0, stream>>>(ws, out);
}